// FilterRois_22703197127187
// MI455X (gfx1250) — compile-verified
//
#include <hip/hip_runtime.h>
#include <hip/hip_bf16.h>
#include <stdint.h>

#define NMS_THRESH   0.7f
#define SCORE_THRESH 0.05f
#define MIN_SIZE     0.001f
#define MAX_PER_IMG  250
#define IMG_W        512.0f
#define IMG_H        512.0f
#define NEGV         (-1e30f)

#define NMAX 2048   // boxes per image (reference: 2048, power of two)
#define NTHR 256    // 8 wave32 per workgroup

// ---------------- gfx1250 async/TDM helpers --------------------------------
typedef int          roi_v4i   __attribute__((ext_vector_type(4)));
typedef unsigned int roi_u32x4 __attribute__((ext_vector_type(4)));
typedef int          roi_i32x8 __attribute__((ext_vector_type(8)));
typedef int          roi_i32x4 __attribute__((ext_vector_type(4)));
typedef __attribute__((address_space(1))) roi_v4i* roi_gptr;
typedef __attribute__((address_space(3))) roi_v4i* roi_lptr;

#if defined(__gfx1250__) && __has_builtin(__builtin_amdgcn_global_load_async_to_lds_b128)
#define ROI_ASYNC_BUILTIN 1
#endif
#if defined(__gfx1250__) && __has_builtin(__builtin_amdgcn_tensor_load_to_lds)
#define ROI_TDM_BUILTIN 1
#endif

__device__ __forceinline__ void roi_async_b128(const float* g, float* l) {
#if defined(ROI_ASYNC_BUILTIN)
  __builtin_amdgcn_global_load_async_to_lds_b128(
      (roi_gptr)g, (roi_lptr)l, /*offset=*/0, /*cpol=*/0);
#else
  uint64_t ga = (uint64_t)(uintptr_t)g;
  uint32_t la = (uint32_t)(uintptr_t)l;
  asm volatile("global_load_async_to_lds_b128 %0, %1, off"
               :: "v"(la), "v"(ga) : "memory");
#endif
}

__device__ __forceinline__ void roi_wait_async0() {
#if __has_builtin(__builtin_amdgcn_s_wait_asynccnt)
  __builtin_amdgcn_s_wait_asynccnt(0);
#else
  asm volatile("s_wait_asynccnt 0" ::: "memory");
#endif
}

// TDM: DMA a run of 4-byte elements from global -> LDS via a D# descriptor.
// Returns true if a TDM op was issued (caller must s_wait_tensorcnt + barrier).
__device__ __forceinline__ bool roi_tdm_load(const float* g, float* l, int nelem4B) {
#if defined(ROI_TDM_BUILTIN)
  uint64_t ga  = (uint64_t)(uintptr_t)g;
  uint32_t lds = (uint32_t)(uintptr_t)l;   // flat-shared low 32 bits == LDS byte addr
  // ---- D# group 0 (CDNA5 ISA) ----
  roi_u32x4 g0;
  g0.x = 1u;                                        // count=1, is_restore=0, gather off
  g0.y = lds;                                       // lds_addr [63:32]
  g0.z = (uint32_t)(ga & 0xFFFFFFFFu);              // global_addr [95:64]
  g0.w = (uint32_t)((ga >> 32) & 0x1FFFFFFu)        // global_addr [120:96]
       | (2u << 30);                                // type=2 ("image")  [127:126]
  // ---- D# group 1 (§8.4): 1-row tile of nelem4B 4-byte elements ----
  uint32_t td0 = (uint32_t)nelem4B;                 // tensor_dim0 (elements)
  roi_i32x8 g1;
  g1[0] = (int)(2u << 16);                          // workgroup_mask=0, data_size=2 (4B)
  g1[1] = (int)((td0 & 0xFFFFu) << 16);             // tensor_dim0[15:0] @ bits 63:48
  g1[2] = (int)(((td0 >> 16) & 0xFFFFu)             // tensor_dim0[31:16]
              | (1u << 16));                        // tensor_dim1 = 1
  g1[3] = (int)((td0 & 0xFFFFu) << 16);             // tile_dim0 = td0 @ bits 127:112
  g1[4] = 1;                                        // tile_dim1 = 1, tile_dim2 = 0
  g1[5] = (int)td0;                                 // tensor_dim0_stride[31:0]
  g1[6] = 0;                                        // stride[47:32] | dim1_stride lo
  g1[7] = 0;                                        // tensor_dim1_stride hi
  roi_i32x4 gz4 = {0, 0, 0, 0};                     // groups 2/3 unused (<=2D)
  roi_i32x8 gz8 = {0, 0, 0, 0, 0, 0, 0, 0};        // 6-arg form's extra group
  __builtin_amdgcn_tensor_load_to_lds(g0, g1, gz4, gz4, gz8, /*cpol=*/0);
  return true;
#else
  (void)g; (void)l; (void)nelem4B;
  return false;
#endif
}

__device__ __forceinline__ void roi_wait_tensor0() {
#if __has_builtin(__builtin_amdgcn_s_wait_tensorcnt)
  __builtin_amdgcn_s_wait_tensorcnt(0);
#else
  asm volatile("s_wait_tensorcnt 0" ::: "memory");
#endif
}

// ---------------------------------------------------------------------------
__global__ __launch_bounds__(NTHR)
void FilterRois_kernel(const float* __restrict__ proposals,  // [P,4] xyxy
                       const float* __restrict__ scores,     // [P]
                       float* __restrict__ out,              // boxes|scores|valid
                       int B, int N) {
  __shared__ __align__(16) float prop[NMAX * 4];  // raw proposals (TDM staged)
  __shared__ __align__(16) float s_key[NMAX];     // raw score -> masked sorted score
  __shared__ int   s_idx[NMAX];
  __shared__ float bx1[NMAX], by1[NMAX], bx2[NMAX], by2[NMAX], ar[NMAX];
  __shared__ unsigned char keep[NMAX];
  __shared__ int   scan[NTHR];
  __shared__ int   tdm_ok;

  const int b    = blockIdx.x;
  const int tid  = threadIdx.x;
  const int base = b * N;

  // ---- stage scores into LDS via async DMA (GLOBAL_LOAD_ASYNC_TO_LDS) ------
  for (int c = tid; c * 4 < N; c += NTHR)
    roi_async_b128(scores + base + c * 4, &s_key[c * 4]);

  // ---- stage proposals (N*16 B) into LDS via Tensor Data Mover -------------
  if ((tid >> 5) == 0) {                    // one wave issues the TDM descriptor
    bool ok = roi_tdm_load(proposals + (size_t)base * 4, prop, N * 4);
    roi_wait_tensor0();                     // this wave's TENSORcnt -> 0
    if (tid == 0) tdm_ok = ok ? 1 : 0;
  }
  roi_wait_async0();
  __syncthreads();                          // LDS (scores + proposals) visible

  if (!tdm_ok) {                            // fallback: cooperative vector copy
    for (int c = tid; c * 4 < N * 4; c += NTHR)
      reinterpret_cast<float4*>(prop)[c] =
          reinterpret_cast<const float4*>(proposals)[base + c];
    __syncthreads();
  }

  // ---- stage 0: sigmoid transform, clip, validity -> masked key ------------
  for (int i = tid; i < N; i += NTHR) {
    float s  = s_key[i];
    float sp = 2.0f / (1.0f + __expf(-s)) - 1.0f;   // sigmoid*2-1
    const float4 p = reinterpret_cast<const float4*>(prop)[i];
    float x1 = fminf(fmaxf(p.x, 0.0f), IMG_W);
    float y1 = fminf(fmaxf(p.y, 0.0f), IMG_H);
    float x2 = fminf(fmaxf(p.z, 0.0f), IMG_W);
    float y2 = fminf(fmaxf(p.w, 0.0f), IMG_H);
    bool valid = (sp >= SCORE_THRESH) && ((x2 - x1) >= MIN_SIZE) && ((y2 - y1) >= MIN_SIZE);
    s_key[i] = valid ? sp : NEGV;
    s_idx[i] = i;
  }
  __syncthreads();

  // ---- bitonic sort: descending score, ties ascending index (stable) -------
  for (int k = 2; k <= N; k <<= 1) {
    for (int j = k >> 1; j > 0; j >>= 1) {
      for (int i = tid; i < N; i += NTHR) {
        int ixj = i ^ j;
        if (ixj > i) {
          float sa = s_key[i];   int ia = s_idx[i];
          float sb = s_key[ixj]; int ib = s_idx[ixj];
          bool bPrecA  = (sb > sa) || (sb == sa && ib < ia);  // descending order
          bool dirAsc  = ((i & k) == 0);
          if (dirAsc == bPrecA) {
            s_key[i] = sb;  s_idx[i] = ib;
            s_key[ixj] = sa; s_idx[ixj] = ia;
          }
        }
      }
      __syncthreads();
    }
  }

  // ---- gather clipped boxes (LDS gather) into sorted SoA, init keep --------
  for (int i = tid; i < N; i += NTHR) {
    int j = s_idx[i];
    const float4 p = reinterpret_cast<const float4*>(prop)[j];
    float x1 = fminf(fmaxf(p.x, 0.0f), IMG_W);
    float y1 = fminf(fmaxf(p.y, 0.0f), IMG_H);
    float x2 = fminf(fmaxf(p.z, 0.0f), IMG_W);
    float y2 = fminf(fmaxf(p.w, 0.0f), IMG_H);
    bx1[i] = x1; by1[i] = y1; bx2[i] = x2; by2[i] = y2;
    ar[i]  = (x2 - x1) * (y2 - y1);
    keep[i] = (s_key[i] >= SCORE_THRESH) ? 1 : 0;   // invalid rows carry NEGV
  }
  __syncthreads();

  // ---- sequential NMS (matches reference scan exactly) ---------------------
  for (int i = 0; i < N - 1; ++i) {
    if (keep[i]) {                       // uniform LDS broadcast read
      float x1 = bx1[i], y1 = by1[i], x2 = bx2[i], y2 = by2[i], ai = ar[i];
      for (int j = i + 1 + tid; j < N; j += NTHR) {
        if (keep[j]) {
          float xx1 = fmaxf(x1, bx1[j]);
          float yy1 = fmaxf(y1, by1[j]);
          float xx2 = fminf(x2, bx2[j]);
          float yy2 = fminf(y2, by2[j]);
          float iw = fmaxf(xx2 - xx1, 0.0f);
          float ih = fmaxf(yy2 - yy1, 0.0f);
          float inter = iw * ih;
          float iou = inter / (ai + ar[j] - inter + 1e-12f);
          if (iou > NMS_THRESH) keep[j] = 0;
        }
      }
    }
    __syncthreads();  // commit this round's suppressions before next i
  }

  // ---- rank kept boxes (block scan), emit top-250 --------------------------
  const int CH = N / NTHR;      // 8 elements per thread
  int cnt = 0;
  for (int c = 0; c < CH; ++c) cnt += keep[tid * CH + c];
  scan[tid] = cnt;
  __syncthreads();
  for (int off = 1; off < NTHR; off <<= 1) {
    int v   = scan[tid];
    int add = (tid >= off) ? scan[tid - off] : 0;
    __syncthreads();
    scan[tid] = v + add;
    __syncthreads();
  }
  int rank = scan[tid] - cnt;   // exclusive prefix over chunks

  float* boxes_out  = out;                                   // [B,250,4]
  float* scores_out = out + (size_t)B * MAX_PER_IMG * 4;     // [B,250]
  float* valid_out  = scores_out + (size_t)B * MAX_PER_IMG;  // [B,250]

  // zero-fill this image's rows (harness poisons d_out)
  for (int r = tid; r < MAX_PER_IMG; r += NTHR) {
    int o = b * MAX_PER_IMG + r;
    boxes_out[o * 4 + 0] = 0.0f;
    boxes_out[o * 4 + 1] = 0.0f;
    boxes_out[o * 4 + 2] = 0.0f;
    boxes_out[o * 4 + 3] = 0.0f;
    scores_out[o] = 0.0f;
    valid_out[o]  = 0.0f;
  }
  __syncthreads();  // workgroup-scope fence: zero-fill before kept writes

  for (int c = 0; c < CH; ++c) {
    int i = tid * CH + c;
    if (keep[i]) {
      if (rank < MAX_PER_IMG) {
        int o = b * MAX_PER_IMG + rank;
        boxes_out[o * 4 + 0] = bx1[i];
        boxes_out[o * 4 + 1] = by1[i];
        boxes_out[o * 4 + 2] = bx2[i];
        boxes_out[o * 4 + 3] = by2[i];
        scores_out[o] = s_key[i];
        valid_out[o]  = 1.0f;
      }
      ++rank;
    }
  }
}

// ---------------------------------------------------------------------------
extern "C" void kernel_launch(void* const* d_in, const int* in_sizes, int n_in,
                              void* d_out, int out_size, void* d_ws, size_t ws_size,
                              hipStream_t stream) {
  (void)n_in; (void)d_ws; (void)ws_size;
  const float* proposals = (const float*)d_in[0];   // [P,4] f32
  const float* scores    = (const float*)d_in[1];   // [P]   f32
  float* out = (float*)d_out;

  // num_images / boxes_per_image are device scalars; derive shapes host-side:
  // out = boxes[B,250,4] ++ scores[B,250] ++ valid[B,250] = 1500*B floats.
  int B = out_size / (MAX_PER_IMG * 6);
  if (B <= 0) B = 1;
  int P = in_sizes[1];
  int N = P / B;
  if (N > NMAX) N = NMAX;

  FilterRois_kernel<<<dim3(B), dim3(NTHR), 0, stream>>>(proposals, scores, out, B, N);
}